// TSFreshFeatureLayer_72773925864082
// MI455X (gfx1250) — compile-verified
//
#include <hip/hip_runtime.h>
#include <math.h>

#define WINDOW 256
#define STRIDE 16
#define NB 64
#define TLEN 8192
#define NF 4
#define NW 497                 // (8192-256)/16 + 1
#define WAVES_PER_BLOCK 8
#define BLOCK (WAVES_PER_BLOCK * 32)

#if __has_builtin(__builtin_amdgcn_global_load_async_to_lds_b128) && \
    __has_builtin(__builtin_amdgcn_s_wait_asynccnt)
#define USE_ASYNC_LDS 1
#else
#define USE_ASYNC_LDS 0
#endif

typedef int v4i __attribute__((ext_vector_type(4)));
typedef __attribute__((address_space(1))) v4i gas_v4i;   // global (AS1) int4
typedef __attribute__((address_space(3))) v4i las_v4i;   // LDS    (AS3) int4

__device__ __forceinline__ float shfl_xor_f(float v, int m) {
  return __shfl_xor(v, m, 32);
}

__global__ __launch_bounds__(BLOCK)
void tsfresh_window_stats(const float* __restrict__ in, float* __restrict__ out) {
  // one wave per (b, window); per-wave LDS tile: 256 timesteps x 4 feats
  __shared__ float buf[WAVES_PER_BLOCK * WINDOW * NF];   // 32 KB

  const int lane = threadIdx.x & 31;
  const int wv   = threadIdx.x >> 5;
  const int task = blockIdx.x * WAVES_PER_BLOCK + wv;
  if (task >= NB * NW) return;           // grid sized exactly; whole-wave guard

  const int b     = task / NW;
  const int w     = task - b * NW;
  const int start = w * STRIDE;

  const float* gsrc  = in + ((size_t)b * TLEN + start) * NF;  // window base
  float*       lbase = &buf[wv * WINDOW * NF];

  // prefetch the tail of the NEXT window for this wave's neighborhood
  __builtin_prefetch(gsrc + (WINDOW + STRIDE) * NF + lane * NF, 0, 1);

  // ---- stage 256x4 floats into LDS: 8 x async b128 per lane, fully coalesced ----
#if USE_ASYNC_LDS
  #pragma unroll
  for (int r = 0; r < 8; ++r) {
    const int e = r * 32 + lane;
    __builtin_amdgcn_global_load_async_to_lds_b128(
        (gas_v4i*)(gsrc + e * NF),
        (las_v4i*)(lbase + e * NF),
        0, 0);
  }
  __builtin_amdgcn_s_wait_asynccnt(0);   // wave-local: LDS tile is private to this wave
#else
  #pragma unroll
  for (int r = 0; r < 8; ++r) {
    const int e = r * 32 + lane;
    const float4 t = *(const float4*)(gsrc + e * NF);
    *(float4*)(lbase + e * NF) = t;      // ds_store_b128; DS ops in-order per wave
  }
#endif

  // ---- per feature: reduce + full 256-element bitonic sort across the wave ----
  #pragma unroll
  for (int f = 0; f < NF; ++f) {
    float v[8];
    #pragma unroll
    for (int r = 0; r < 8; ++r)
      v[r] = lbase[(r * 32 + lane) * NF + f];

    // mean and E[x^2] via butterfly reduction (pairs batched per step for ILP)
    float s = 0.f, s2 = 0.f;
    #pragma unroll
    for (int r = 0; r < 8; ++r) { s += v[r]; s2 += v[r] * v[r]; }
    #pragma unroll
    for (int off = 16; off >= 1; off >>= 1) {
      const float a0 = shfl_xor_f(s,  off);
      const float a1 = shfl_xor_f(s2, off);
      s  += a0;
      s2 += a1;
    }
    const float mean = s * (1.f / 256.f);
    float var = s2 * (1.f / 256.f) - mean * mean;
    var = var > 0.f ? var : 0.f;
    const float stdv = sqrtf(var);

    // counts above / below mean (order-independent, use current multiset)
    float ca = 0.f, cb = 0.f;
    #pragma unroll
    for (int r = 0; r < 8; ++r) {
      ca += (v[r] > mean) ? 1.f : 0.f;
      cb += (v[r] < mean) ? 1.f : 0.f;
    }
    #pragma unroll
    for (int off = 16; off >= 1; off >>= 1) {
      const float a0 = shfl_xor_f(ca, off);
      const float a1 = shfl_xor_f(cb, off);
      ca += a0;
      cb += a1;
    }

    // ---- bitonic sort, ascending in e = r*32 + lane ----
    #pragma unroll
    for (int k = 2; k <= 256; k <<= 1) {
      // register-distance stages (j >= 32): partner is r ^ (j>>5), same lane
      #pragma unroll
      for (int j = 128; j >= 32; j >>= 1) {
        if (j < k) {
          const int rd = j >> 5;
          #pragma unroll
          for (int r = 0; r < 8; ++r) {
            const int pr = r ^ rd;
            if (pr > r) {
              const bool asc = (((r * 32) & k) == 0);  // same for r and pr
              const float lo = fminf(v[r], v[pr]);
              const float hi = fmaxf(v[r], v[pr]);
              v[r]  = asc ? lo : hi;
              v[pr] = asc ? hi : lo;
            }
          }
        }
      }
      // lane-distance stages (j <= 16): issue all 8 shuffles, then all selects,
      // so the scheduler can batch ds ops and drain with partial dscnt waits
      #pragma unroll
      for (int j = 16; j >= 1; j >>= 1) {
        if (j < k) {
          float p[8];
          #pragma unroll
          for (int r = 0; r < 8; ++r)
            p[r] = shfl_xor_f(v[r], j);
          #pragma unroll
          for (int r = 0; r < 8; ++r) {
            const int   e = r * 32 + lane;
            const bool  asc     = ((e & k) == 0);
            const bool  lower   = ((lane & j) == 0);
            const bool  keepMin = (asc == lower);
            const float lo = fminf(v[r], p[r]);
            const float hi = fmaxf(v[r], p[r]);
            v[r] = keepMin ? lo : hi;
          }
        }
      }
    }

    // ---- exact order statistics from fixed (reg, lane) slots ----
    const float mn   = __shfl(v[0], 0, 32);
    const float s63  = __shfl(v[1], 31, 32);
    const float s64  = __shfl(v[2], 0, 32);
    const float s127 = __shfl(v[3], 31, 32);
    const float s128 = __shfl(v[4], 0, 32);
    const float s191 = __shfl(v[5], 31, 32);
    const float s192 = __shfl(v[6], 0, 32);
    const float mx   = __shfl(v[7], 31, 32);

    const float q25 = s63 + 0.75f * (s64 - s63);    // (n-1)*0.25 = 63.75
    const float med = 0.5f * (s127 + s128);         // 127.5
    const float q75 = s191 + 0.25f * (s192 - s191); // 191.25
    const float iqr = q75 - q25;

    if (lane == 0) {
      float* o = out + (size_t)task * (NF * 8) + f * 8;   // [mean,std,mn,mx,med,iqr,above,below]
      *(float4*)(o + 0) = make_float4(mean, stdv, mn, mx);
      *(float4*)(o + 4) = make_float4(med, iqr, ca, cb);
    }
  }
}

extern "C" void kernel_launch(void* const* d_in, const int* in_sizes, int n_in,
                              void* d_out, int out_size, void* d_ws, size_t ws_size,
                              hipStream_t stream) {
  (void)in_sizes; (void)n_in; (void)out_size; (void)d_ws; (void)ws_size;
  const float* in  = (const float*)d_in[0];
  float*       out = (float*)d_out;
  const int tasks  = NB * NW;                               // 31808 waves
  const int blocks = (tasks + WAVES_PER_BLOCK - 1) / WAVES_PER_BLOCK;  // 3976
  tsfresh_window_stats<<<blocks, BLOCK, 0, stream>>>(in, out);
}